// CrossLastLayer_77111842832928
// MI455X (gfx1250) — compile-verified
//
#include <hip/hip_runtime.h>
#include <hip/hip_bf16.h>

// ---------------------------------------------------------------------------
// CrossLastLayer on MI455X (gfx1250, wave32, WMMA).
// Bandwidth-bound (~3-4 GB sparse traffic vs 46 GFLOP fp32 GEMM): keep fp32,
// V_WMMA_F32_16X16X4_F32 for GEMMs, async global->LDS staging, prefetched
// wave-per-edge SpMM with f32 atomics.
// ---------------------------------------------------------------------------

#define FDIM   128
#define ALPHA  0.2f
#define RATE   0.5f

typedef float v2f __attribute__((ext_vector_type(2)));
typedef float v8f __attribute__((ext_vector_type(8)));
typedef int   v4i __attribute__((ext_vector_type(4)));

#if defined(__gfx1250__) && __has_builtin(__builtin_amdgcn_global_load_async_to_lds_b128) && \
    __has_builtin(__builtin_amdgcn_s_wait_asynccnt)
#define HAS_ASYNC_LDS 1
typedef __attribute__((address_space(1))) v4i* gptr_v4i;
typedef __attribute__((address_space(3))) v4i* lptr_v4i;
#else
#define HAS_ASYNC_LDS 0
#endif

// ---------------------------------------------------------------------------
// GEMM: out[row, 0:128] (+)= scale * ( act( sum_seg A_seg @ W_seg + bias ) )
//   Block = 256 threads (8 waves); tile 128 rows x 128 cols; K step 4 via
//   V_WMMA_F32_16X16X4_F32.
// LDS layouts:
//   sA : row-major, stride 132 floats (pad 4) -> A-frag ds_load_b64 hits
//        rotating banks (stride 132 ≡ 4 banks/row).
//   sWp: K-pair-major float2 [64][128]; B-frag (W[k..k+1][n] or W[k+2..k+3][n])
//        is one contiguous ds_load_b64 into an even VGPR pair. No shuffle movs.
// Fragment layouts per CDNA5 ISA 7.12.2 (f32 16x16x4):
//   A: v0/v1 = K{0,1} (lanes 0-15) / K{2,3} (lanes 16-31), M = lane%16
//   B: v0/v1 = K{0,1} (lanes 0-15) / K{2,3} (lanes 16-31), N = lane%16
//   C/D: vr -> row r + 8*(lane/16), col lane%16
// ---------------------------------------------------------------------------
__global__ __launch_bounds__(256)
void gemm128_wmma(const float* __restrict__ A0,
                  const float* __restrict__ A1,     // second concat segment (or null)
                  const float* __restrict__ W,      // [nseg*128, 128]
                  const float* __restrict__ bias,   // [128] or null
                  float* __restrict__ out,          // [Nrows, 128]
                  int Nrows, int nseg, int leaky, float scale, int accum)
{
    constexpr int LDA = FDIM + 4;                       // padded A stride
    __shared__ __align__(16) float  sA[FDIM * LDA];     // ~67.6 KB
    __shared__ __align__(16) float2 sWp[(FDIM / 2) * FDIM]; // 64 KB: [kpair][n]

    const int tid  = threadIdx.x;
    const int wave = tid >> 5;
    const int lane = tid & 31;
    const int l    = lane & 15;
    const int hi   = lane >> 4;          // K-half within fragment
    const int rowBase  = blockIdx.x * FDIM;
    const int m0       = wave * 16;
    const bool fullTile = (rowBase + FDIM) <= Nrows;

    v8f acc[8];
#pragma unroll
    for (int nt = 0; nt < 8; ++nt) acc[nt] = (v8f)0.0f;

    for (int seg = 0; seg < nseg; ++seg) {
        const float* __restrict__ A  = (seg == 0) ? A0 : A1;
        const float* __restrict__ Ws = W + (size_t)seg * FDIM * FDIM;

        // --- stage W as K-pairs: sWp[p][n] = {W[2p][n], W[2p+1][n]} ---------
        for (int i = tid; i < (FDIM / 2) * FDIM; i += 256) {
            const int p = i >> 7;              // 0..63
            const int n = i & (FDIM - 1);      // 0..127
            sWp[i] = make_float2(Ws[(size_t)(2 * p) * FDIM + n],
                                 Ws[(size_t)(2 * p + 1) * FDIM + n]);
        }

        // --- stage A tile (async global->LDS on the full-tile fast path) ----
        if (fullTile) {
#if HAS_ASYNC_LDS
            for (int i = tid; i < FDIM * (FDIM / 4); i += 256) {
                const int r  = i >> 5;
                const int c4 = (i & 31) * 4;
                __builtin_amdgcn_global_load_async_to_lds_b128(
                    (gptr_v4i)&A[(size_t)(rowBase + r) * FDIM + c4],
                    (lptr_v4i)&sA[r * LDA + c4],
                    0, 0);
            }
            __builtin_amdgcn_s_wait_asynccnt(0);
#else
            for (int i = tid; i < FDIM * (FDIM / 4); i += 256) {
                const int r  = i >> 5;
                const int c4 = (i & 31) * 4;
                *(float4*)&sA[r * LDA + c4] =
                    *(const float4*)&A[(size_t)(rowBase + r) * FDIM + c4];
            }
#endif
        } else {
            for (int i = tid; i < FDIM * (FDIM / 4); i += 256) {
                const int r  = i >> 5;
                const int c4 = (i & 31) * 4;
                const int grow = rowBase + r;
                float4 va = make_float4(0.f, 0.f, 0.f, 0.f);
                if (grow < Nrows)
                    va = *(const float4*)&A[(size_t)grow * FDIM + c4];
                *(float4*)&sA[r * LDA + c4] = va;
            }
        }
        __syncthreads();

        // --- K loop: 32 steps of 4, 8 column tiles per step -----------------
        for (int k = 0; k < FDIM; k += 4) {
            const int kk = k + 2 * hi;
            const float2 af = *(const float2*)&sA[(m0 + l) * LDA + kk];
            v2f a; a.x = af.x; a.y = af.y;
            const int prow = ((k >> 1) + hi) * FDIM + l;
#pragma unroll
            for (int nt = 0; nt < 8; ++nt) {
                const float2 bf = sWp[prow + nt * 16];
                v2f b; b.x = bf.x; b.y = bf.y;
                acc[nt] = __builtin_amdgcn_wmma_f32_16x16x4_f32(
                    /*neg_a=*/false, a, /*neg_b=*/false, b,
                    /*c_mod=*/(short)0, acc[nt],
                    /*reuse_a=*/false, /*reuse_b=*/false);
            }
        }
        __syncthreads();
    }

    // --- epilogue: bias -> optional LeakyReLU -> scale -> store/accumulate --
#pragma unroll
    for (int nt = 0; nt < 8; ++nt) {
        const int col = nt * 16 + l;
        const float bval = (bias != nullptr) ? bias[col] : 0.0f;
#pragma unroll
        for (int r = 0; r < 8; ++r) {
            const int row = rowBase + m0 + r + 8 * hi;
            if (row < Nrows) {
                float v = acc[nt][r] + bval;
                if (leaky) v = (v > 0.0f) ? v : ALPHA * v;
                v *= scale;
                float* p = &out[(size_t)row * FDIM + col];
                if (accum) *p += v; else *p = v;
            }
        }
    }
}

// ---------------------------------------------------------------------------
// SpMM (COO, D=128): one wave per EPW edges. Lane i owns cols 4i..4i+3.
// Prefetch all gathered rows (global_prefetch_b8), then gather/scale/scatter
// with f32 atomics. Each gather is a fully-coalesced 512B row.
// ---------------------------------------------------------------------------
#define EPW 4

__global__ __launch_bounds__(256)
void spmm_atomic(const int* __restrict__ row, const int* __restrict__ col,
                 const float* __restrict__ val,
                 const float* __restrict__ x,   // [n_in, 128]
                 float* __restrict__ out,       // [n_out, 128] pre-zeroed
                 int E)
{
    const int gid  = blockIdx.x * blockDim.x + threadIdx.x;
    const int wv   = gid >> 5;
    const int lane = gid & 31;
    const int base = wv * EPW;

#pragma unroll
    for (int j = 0; j < EPW; ++j) {
        const int e = base + j;
        if (e < E) {
            const int c = col[e];
            __builtin_prefetch(&x[(size_t)c * FDIM + lane * 4], 0, 0);
        }
    }
#pragma unroll
    for (int j = 0; j < EPW; ++j) {
        const int e = base + j;
        if (e >= E) continue;
        const int   r = row[e];
        const int   c = col[e];
        const float v = val[e];
        const float4 xv = *(const float4*)&x[(size_t)c * FDIM + lane * 4];
        float* o = &out[(size_t)r * FDIM + lane * 4];
        atomicAdd(o + 0, v * xv.x);
        atomicAdd(o + 1, v * xv.y);
        atomicAdd(o + 2, v * xv.z);
        atomicAdd(o + 3, v * xv.w);
    }
}

// ---------------------------------------------------------------------------
// Zero fill (float4 granularity).
// ---------------------------------------------------------------------------
__global__ __launch_bounds__(256)
void zero4(float* __restrict__ buf, int total4)
{
    const int i = blockIdx.x * blockDim.x + threadIdx.x;
    if (i < total4) ((float4*)buf)[i] = make_float4(0.f, 0.f, 0.f, 0.f);
}

// ---------------------------------------------------------------------------
// In-place bias + LeakyReLU after SpMM accumulation. One float4 per thread.
// ---------------------------------------------------------------------------
__global__ __launch_bounds__(256)
void bias_leaky(float* __restrict__ buf, const float* __restrict__ bias, int total4)
{
    const int i = blockIdx.x * blockDim.x + threadIdx.x;
    if (i >= total4) return;
    const int c4 = (i & 31) * 4;
    float4 v = ((float4*)buf)[i];
    v.x += bias[c4 + 0]; v.y += bias[c4 + 1];
    v.z += bias[c4 + 2]; v.w += bias[c4 + 3];
    v.x = (v.x > 0.f) ? v.x : ALPHA * v.x;
    v.y = (v.y > 0.f) ? v.y : ALPHA * v.y;
    v.z = (v.z > 0.f) ? v.z : ALPHA * v.z;
    v.w = (v.w > 0.f) ? v.w : ALPHA * v.w;
    ((float4*)buf)[i] = v;
}

// ---------------------------------------------------------------------------
// Orchestration.
// ---------------------------------------------------------------------------
extern "C" void kernel_launch(void* const* d_in, const int* in_sizes, int n_in,
                              void* d_out, int out_size, void* d_ws, size_t ws_size,
                              hipStream_t stream)
{
    const float* src_ufea = (const float*)d_in[0];
    const float* tgt_ufea = (const float*)d_in[1];
    const int*   s_uv_row = (const int*)d_in[2];
    const int*   s_uv_col = (const int*)d_in[3];
    const float* s_uv_val = (const float*)d_in[4];
    const int*   s_vu_row = (const int*)d_in[5];
    const int*   s_vu_col = (const int*)d_in[6];
    const float* s_vu_val = (const float*)d_in[7];
    const int*   t_uv_row = (const int*)d_in[8];
    const int*   t_uv_col = (const int*)d_in[9];
    const float* t_uv_val = (const float*)d_in[10];
    const int*   t_vu_row = (const int*)d_in[11];
    const int*   t_vu_col = (const int*)d_in[12];
    const float* t_vu_val = (const float*)d_in[13];
    const float* w_gc1  = (const float*)d_in[14]; const float* b_gc1  = (const float*)d_in[15];
    const float* w_gc3m = (const float*)d_in[16]; const float* b_gc3m = (const float*)d_in[17];
    const float* w_gc3s = (const float*)d_in[18]; const float* b_gc3s = (const float*)d_in[19];
    const float* w_gc2  = (const float*)d_in[20]; const float* b_gc2  = (const float*)d_in[21];
    const float* w_gc4m = (const float*)d_in[22]; const float* b_gc4m = (const float*)d_in[23];
    const float* w_gc4s = (const float*)d_in[24]; const float* b_gc4s = (const float*)d_in[25];
    const float* w_sum  = (const float*)d_in[26]; const float* b_sum  = (const float*)d_in[27];
    const float* w_sus  = (const float*)d_in[28]; const float* b_sus  = (const float*)d_in[29];
    const float* w_tum  = (const float*)d_in[30]; const float* b_tum  = (const float*)d_in[31];
    const float* w_tus  = (const float*)d_in[32]; const float* b_tus  = (const float*)d_in[33];

    const int NU = in_sizes[0] / FDIM;     // 100000 (== NI here)
    const int E  = in_sizes[2];            // 640000

    // Scratch: three [N, 128] fp32 buffers (51.2 MB each).
    const size_t NUD = (size_t)NU * FDIM;
    float* B0 = (float*)d_ws;              // dense GEMM result (SpMM input)
    float* B1 = B0 + NUD;                  // hidden layer (ho)
    float* B2 = B1 + NUD;                  // ho_mean / ho_logstd

    float* out_mean   = (float*)d_out;
    float* out_logstd = out_mean + NUD;

    const dim3 blk(256);
    const dim3 gGemm((NU + FDIM - 1) / FDIM);                   // 782 blocks
    const int  nWaves = (E + EPW - 1) / EPW;                    // wave per EPW edges
    const dim3 gSpmm((nWaves * 32 + 255) / 256);
    const int  total4 = (int)(NUD / 4);
    const dim3 gElem((total4 + 255) / 256);

    struct Branch {
        const float* ufea;
        const int *vu_row, *vu_col; const float* vu_val;
        const int *uv_row, *uv_col; const float* uv_val;
        const float *w1, *b1, *wm, *bm, *ws_, *bs_;
        const float *w_um, *b_um, *w_us, *b_us;
        int accum;
    } br[2] = {
        { src_ufea, s_vu_row, s_vu_col, s_vu_val, s_uv_row, s_uv_col, s_uv_val,
          w_gc1, b_gc1, w_gc3m, b_gc3m, w_gc3s, b_gc3s,
          w_sum, b_sum, w_sus, b_sus, 0 },
        { tgt_ufea, t_vu_row, t_vu_col, t_vu_val, t_uv_row, t_uv_col, t_uv_val,
          w_gc2, b_gc2, w_gc4m, b_gc4m, w_gc4s, b_gc4s,
          w_tum, b_tum, w_tus, b_tus, 1 },
    };

    for (int i = 0; i < 2; ++i) {
        const Branch& b = br[i];
        // gc first hop: ho = leaky( spmm(vu, ufea @ W1) + b1 )   [NI, D]
        gemm128_wmma<<<gGemm, blk, 0, stream>>>(b.ufea, nullptr, b.w1, nullptr,
                                                B0, NU, 1, 0, 1.0f, 0);
        zero4<<<gElem, blk, 0, stream>>>(B1, total4);
        spmm_atomic<<<gSpmm, blk, 0, stream>>>(b.vu_row, b.vu_col, b.vu_val, B0, B1, E);
        bias_leaky<<<gElem, blk, 0, stream>>>(B1, b.b1, total4);

        // mean head: ho_mean = leaky( spmm(uv, ho @ Wm) + bm )   [NU, D]
        gemm128_wmma<<<gGemm, blk, 0, stream>>>(B1, nullptr, b.wm, nullptr,
                                                B0, NU, 1, 0, 1.0f, 0);
        zero4<<<gElem, blk, 0, stream>>>(B2, total4);
        spmm_atomic<<<gSpmm, blk, 0, stream>>>(b.uv_row, b.uv_col, b.uv_val, B0, B2, E);
        bias_leaky<<<gElem, blk, 0, stream>>>(B2, b.bm, total4);
        // union: out_mean (+)= RATE * (concat(ho_mean, ufea) @ W_um + b_um)
        gemm128_wmma<<<gGemm, blk, 0, stream>>>(B2, b.ufea, b.w_um, b.b_um,
                                                out_mean, NU, 2, 0, RATE, b.accum);

        // logstd head (B2 reused)
        gemm128_wmma<<<gGemm, blk, 0, stream>>>(B1, nullptr, b.ws_, nullptr,
                                                B0, NU, 1, 0, 1.0f, 0);
        zero4<<<gElem, blk, 0, stream>>>(B2, total4);
        spmm_atomic<<<gSpmm, blk, 0, stream>>>(b.uv_row, b.uv_col, b.uv_val, B0, B2, E);
        bias_leaky<<<gElem, blk, 0, stream>>>(B2, b.bs_, total4);
        gemm128_wmma<<<gGemm, blk, 0, stream>>>(B2, b.ufea, b.w_us, b.b_us,
                                                out_logstd, NU, 2, 0, RATE, b.accum);
    }
}